// ListNetLoss_33904471835359
// MI455X (gfx1250) — compile-verified
//
#include <hip/hip_runtime.h>
#include <hip/hip_bf16.h>

#define BDIM 256
#define WAVES_PER_BLOCK (BDIM / 32)

constexpr int Bn = 8192;   // batch
constexpr int Nn = 4096;   // num items
constexpr int Ln = 512;    // list length
#define NEGV (-1e30f)

typedef __attribute__((ext_vector_type(2))) float v2f;
typedef __attribute__((ext_vector_type(8))) float v8f;

// logsumexp pair combine: (m,s) <- (m,s) (+) (m2,s2)
__device__ __forceinline__ void lse_combine(float& m, float& s, float m2, float s2) {
    float M = fmaxf(m, m2);
    s = s * expf(m - M) + s2 * expf(m2 - M);
    m = M;
}

// One wave32 per row. Lane owns 16 contiguous elements.
// Writes per-row sum of ll = g - log_denom (valid entries only) to row_out[row].
__global__ __launch_bounds__(BDIM) void pl_row_kernel(const float* __restrict__ scores,
                                                      const int*   __restrict__ seq,
                                                      float*       __restrict__ row_out) {
    const int lane = threadIdx.x & 31;
    const int wave = threadIdx.x >> 5;
    const int row  = blockIdx.x * WAVES_PER_BLOCK + wave;

    const long rb = (long)row * Ln + lane * 16;
    const int4* sp = (const int4*)(seq + rb);       // 64B-aligned coalesced index loads
    int idxs[16];
#pragma unroll
    for (int q = 0; q < 4; ++q) {
        int4 v = sp[q];
        idxs[q * 4 + 0] = v.x; idxs[q * 4 + 1] = v.y;
        idxs[q * 4 + 2] = v.z; idxs[q * 4 + 3] = v.w;
    }

    const float* __restrict__ srow = scores + (long)row * Nn;
    float gv[16];
    unsigned vmask = 0u;
#pragma unroll
    for (int t = 0; t < 16; ++t) {
        int id = idxs[t];
        bool valid = (id != -1);
        int safe = valid ? id : 0;          // reference: where(mask, y, 0) -> gather -> mask
        float g = srow[safe];
        gv[t] = valid ? g : NEGV;
        vmask |= (valid ? 1u : 0u) << t;
    }

    // In-chunk reverse inclusive suffix scan (t = 15..0), keep per-element (m,s).
    float m = NEGV, s = 0.0f;               // identity
    float msuf[16], ssuf[16];
#pragma unroll
    for (int t = 15; t >= 0; --t) {
        float M = fmaxf(m, gv[t]);
        s = s * expf(m - M) + expf(gv[t] - M);
        m = M;
        msuf[t] = m; ssuf[t] = s;
    }

    // Cross-lane inclusive suffix scan of chunk aggregates (m,s), then exclusive shift.
    float am = m, as = s;
#pragma unroll
    for (int off = 1; off < 32; off <<= 1) {
        float om = __shfl_down(am, off, 32);
        float os = __shfl_down(as, off, 32);
        if (lane + off < 32) lse_combine(am, as, om, os);
    }
    float rm = __shfl_down(am, 1, 32);      // aggregate of lanes [lane+1 .. 31]
    float rs = __shfl_down(as, 1, 32);
    if (lane == 31) { rm = NEGV; rs = 0.0f; }

    // Fuse: log_denom[t] = LSE(chunk_suffix[t] (+) right_exclusive)
    float acc = 0.0f;
#pragma unroll
    for (int t = 0; t < 16; ++t) {
        float M = fmaxf(msuf[t], rm);
        float S = ssuf[t] * expf(msuf[t] - M) + rs * expf(rm - M);
        float ld = M + logf(S);
        if (vmask & (1u << t)) acc += gv[t] - ld;
    }

    // Deterministic wave sum.
#pragma unroll
    for (int off = 16; off >= 1; off >>= 1) acc += __shfl_xor(acc, off, 32);
    if (lane == 0) row_out[row] = acc;
}

// Single block: deterministic reduction of 8192 per-row sums -> -mean.
// Intra-wave sum uses V_WMMA_F32_16X16X4_F32 (f32 WMMA, exact f32 math):
// A[16x4] holds the 32 lane values (K=0 cols per half-wave), B = ones(4x16),
// so D[m,n] = val[m] + val[m+16]; summing D column 0 gives the wave total.
__global__ __launch_bounds__(BDIM) void pl_reduce_kernel(const float* __restrict__ row_ll,
                                                         float* __restrict__ out) {
    const int tid  = threadIdx.x;
    const int lane = tid & 31;
    const int wave = tid >> 5;

    float v = 0.0f;
#pragma unroll
    for (int k = 0; k < Bn / BDIM; ++k) v += row_ll[tid + k * BDIM];

    __shared__ float red[WAVES_PER_BLOCK];

#if __has_builtin(__builtin_amdgcn_wmma_f32_16x16x4_f32)
    v2f a; a[0] = v;    a[1] = 0.0f;   // A: lanes 0-15 -> K=0,1 ; lanes 16-31 -> K=2,3
    v2f b; b[0] = 1.0f; b[1] = 1.0f;   // B: all ones
    v8f c = {};
    c = __builtin_amdgcn_wmma_f32_16x16x4_f32(false, a, false, b, (short)0, c, false, false);
    // lane n holds D[0..7, n], lane n+16 holds D[8..15, n]
    float part = c[0] + c[1] + c[2] + c[3] + c[4] + c[5] + c[6] + c[7];
    float tot = part + __shfl_down(part, 16, 32);   // lane 0: full wave sum
#else
    float tot = v;
#pragma unroll
    for (int off = 16; off >= 1; off >>= 1) tot += __shfl_xor(tot, off, 32);
#endif

    if (lane == 0) red[wave] = tot;
    __syncthreads();
    if (tid == 0) {
        float t = 0.0f;
#pragma unroll
        for (int w = 0; w < WAVES_PER_BLOCK; ++w) t += red[w];
        out[0] = -t / (float)Bn;
    }
}

extern "C" void kernel_launch(void* const* d_in, const int* in_sizes, int n_in,
                              void* d_out, int out_size, void* d_ws, size_t ws_size,
                              hipStream_t stream) {
    const float* scores = (const float*)d_in[0];   // [B, N] f32
    const int*   seq    = (const int*)d_in[1];     // [B, L] int (padded with -1)
    float* rowbuf = (float*)d_ws;                  // B floats of scratch

    dim3 grid(Bn / WAVES_PER_BLOCK);
    pl_row_kernel<<<grid, BDIM, 0, stream>>>(scores, seq, rowbuf);
    pl_reduce_kernel<<<1, BDIM, 0, stream>>>(rowbuf, (float*)d_out);
}